// GNN_39213051412908
// MI455X (gfx1250) — compile-verified
//
#include <hip/hip_runtime.h>

#define N_NODES 100000
#define N_EDGES 600000
#define D 128
#define LDS_STRIDE 130  // padded (even -> 8B-aligned b64 pairs, low bank conflicts)

typedef __attribute__((ext_vector_type(2))) float v2f;
typedef __attribute__((ext_vector_type(8))) float v8f;

// ---------------------------------------------------------------------------
// Degree / normalization kernels
// ---------------------------------------------------------------------------
__global__ __launch_bounds__(256) void deg_init_kernel(float* __restrict__ deg) {
    int i = blockIdx.x * 256 + threadIdx.x;
    if (i < N_NODES) deg[i] = 1.0f;  // self-loop contributes 1 to every node
}

__global__ __launch_bounds__(256) void deg_count_kernel(const int* __restrict__ ei,
                                                        float* __restrict__ deg) {
    int e = blockIdx.x * 256 + threadIdx.x;
    if (e < N_EDGES) {
        int d = ei[N_EDGES + e];  // edge_index[1][e]
        unsafeAtomicAdd(&deg[d], 1.0f);
    }
}

__global__ __launch_bounds__(256) void deg_to_dinv_kernel(float* __restrict__ deg) {
    int i = blockIdx.x * 256 + threadIdx.x;
    if (i < N_NODES) deg[i] = rsqrtf(deg[i]);  // deg >= 1 always (self loops)
}

// ---------------------------------------------------------------------------
// Dense GEMM  H[M,128] = (RELU? max(A,0):A)[M,128] @ W[128,128]
// fp32 WMMA 16x16x4. Block = 256 threads = 8 waves; block owns 16 rows and
// covers all 128 columns (one 16x16 tile per wave). W is staged once per
// block into LDS, TRANSPOSED with stride 130, so the per-iteration B fragment
// (the k,k+1 pair for column n) is one aligned ds_load_b64.
//
// A-frag (16x4 f32, ISA 7.12.2): lane%16 = M row; lanes 0-15 hold K={0,1},
// lanes 16-31 hold K={2,3} (contiguous pair -> one global_load_b64).
// B-frag (4x16 f32): lane%16 = N col; lanes 0-15 K={0,1}, 16-31 K={2,3}.
// C/D  (16x16 f32): VGPR r: lanes 0-15 -> M=r, lanes 16-31 -> M=r+8.
//
// Out-of-range rows are handled by clamping the row pointer (loads stay
// unconditional so EXEC stays all-ones and the loop has no branches); the
// bogus results only land in discarded output rows, guarded at the store.
// ---------------------------------------------------------------------------
template <bool RELU>
__global__ __launch_bounds__(256) void gcn_gemm_wmma(const float* __restrict__ A,
                                                     const float* __restrict__ W,
                                                     float* __restrict__ H, int M) {
    __shared__ float sW[D * LDS_STRIDE];  // W transposed: sW[n*130 + k]

    // --- stage W into LDS (coalesced b128 reads, scattered ds stores) ------
    for (int i = threadIdx.x * 4; i < D * D; i += 256 * 4) {
        const int k = i >> 7;        // row of W
        const int n = i & 127;       // col of W (multiple of 4)
        const float4 w4 = *(const float4*)(W + i);
        sW[(n + 0) * LDS_STRIDE + k] = w4.x;
        sW[(n + 1) * LDS_STRIDE + k] = w4.y;
        sW[(n + 2) * LDS_STRIDE + k] = w4.z;
        sW[(n + 3) * LDS_STRIDE + k] = w4.w;
    }
    __syncthreads();

    const int lane = threadIdx.x & 31;
    const int wv   = threadIdx.x >> 5;          // 0..7 -> which 16-col strip
    const int m0   = blockIdx.x << 4;           // 16 rows per block
    const int row  = m0 + (lane & 15);
    const int hi   = lane >> 4;                 // 0 or 1 (K pair select)
    const int ncol = (wv << 4) + (lane & 15);
    const float* arow = A + (size_t)(row < M ? row : 0) * D;  // clamped
    const float* brow = sW + ncol * LDS_STRIDE + (hi << 1);

    v8f c = {};
#pragma unroll
    for (int kb = 0; kb < D; kb += 4) {
        v2f a = *(const v2f*)(arow + kb + (hi << 1));  // global_load_b64
        if (RELU) {
            a.x = a.x > 0.0f ? a.x : 0.0f;
            a.y = a.y > 0.0f ? a.y : 0.0f;
        }
        const v2f b = *(const v2f*)(brow + kb);        // ds_load_b64
        c = __builtin_amdgcn_wmma_f32_16x16x4_f32(false, a, false, b,
                                                  (short)0, c, false, false);
    }

    const int mbase = m0 + (hi << 3);
#pragma unroll
    for (int r = 0; r < 8; ++r) {
        const int m = mbase + r;
        if (m < M) H[(size_t)m * D + ncol] = c[r];
    }
}

// ---------------------------------------------------------------------------
// out[i,:] = dinv[i]^2 * h[i,:] + bias   (fused self-loop message + bias)
// one thread per float4 chunk; 32 chunks per node
// ---------------------------------------------------------------------------
__global__ __launch_bounds__(256) void agg_init_kernel(const float* __restrict__ h,
                                                       const float* __restrict__ dinv,
                                                       const float* __restrict__ bias,
                                                       float* __restrict__ out) {
    size_t idx = (size_t)blockIdx.x * 256 + threadIdx.x;   // over N*D/4 float4s
    if (idx >= (size_t)N_NODES * (D / 4)) return;
    const int node = (int)(idx >> 5);                      // D/4 == 32 chunks/node
    const int j    = (int)(idx & 31) << 2;
    float s = dinv[node];
    s = s * s;
    const float4 hv = *(const float4*)(h + (size_t)node * D + j);
    const float4 bv = *(const float4*)(bias + j);
    float4 o;
    o.x = s * hv.x + bv.x;
    o.y = s * hv.y + bv.y;
    o.z = s * hv.z + bv.z;
    o.w = s * hv.w + bv.w;
    *(float4*)(out + (size_t)node * D + j) = o;
}

// ---------------------------------------------------------------------------
// Edge scatter: one wave per edge. Lanes gather a coalesced 512B row of
// h[src], scale by dinv[src]*dinv[dst] (scalarized via readfirstlane since
// the edge id is wave-uniform), and scatter with hardware f32 atomics that
// resolve in L2 (the 51MB destination buffer is L2-resident on a 192MB L2).
// ---------------------------------------------------------------------------
__global__ __launch_bounds__(256) void agg_edges_kernel(const int* __restrict__ ei,
                                                        const float* __restrict__ dinv,
                                                        const float* __restrict__ h,
                                                        float* __restrict__ out) {
    int tid = blockIdx.x * 256 + threadIdx.x;
    int e = tid >> 5;
    if (e >= N_EDGES) return;
    const int lane = tid & 31;
    int s = __builtin_amdgcn_readfirstlane(ei[e]);             // wave-uniform
    int d = __builtin_amdgcn_readfirstlane(ei[N_EDGES + e]);   // wave-uniform
    const float nrm = dinv[s] * dinv[d];
    const float4 hv = *(const float4*)(h + (size_t)s * D + (lane << 2));
    float* op = out + (size_t)d * D + (lane << 2);
    unsafeAtomicAdd(op + 0, nrm * hv.x);
    unsafeAtomicAdd(op + 1, nrm * hv.y);
    unsafeAtomicAdd(op + 2, nrm * hv.z);
    unsafeAtomicAdd(op + 3, nrm * hv.w);
}

// ---------------------------------------------------------------------------
extern "C" void kernel_launch(void* const* d_in, const int* in_sizes, int n_in,
                              void* d_out, int out_size, void* d_ws, size_t ws_size,
                              hipStream_t stream) {
    const float* x  = (const float*)d_in[0];
    const int*   ei = (const int*)d_in[1];   // [2, N_EDGES] int32
    const float* W1 = (const float*)d_in[2];
    const float* b1 = (const float*)d_in[3];
    const float* W2 = (const float*)d_in[4];
    const float* b2 = (const float*)d_in[5];
    float* out = (float*)d_out;

    char* ws = (char*)d_ws;
    float* dinv = (float*)ws;                                   // N floats
    size_t off = ((size_t)N_NODES * 4 + 511) & ~(size_t)511;
    float* hbuf = (float*)(ws + off);                           // N*D (h1 then h2)
    float* abuf = (float*)(ws + off + (size_t)N_NODES * D * 4); // N*D (aggregated)

    const int nb_nodes   = (N_NODES + 255) / 256;
    const int nb_edges   = (N_EDGES + 255) / 256;
    const int nb_gemm    = (N_NODES + 15) / 16;
    const int nb_init    = (int)(((size_t)N_NODES * (D / 4) + 255) / 256);
    const int nb_scatter = (int)(((size_t)N_EDGES * 32 + 255) / 256);

    // symmetric degree normalization
    deg_init_kernel<<<nb_nodes, 256, 0, stream>>>(dinv);
    deg_count_kernel<<<nb_edges, 256, 0, stream>>>(ei, dinv);
    deg_to_dinv_kernel<<<nb_nodes, 256, 0, stream>>>(dinv);

    // layer 1: h1 = x @ W1 ; out1 = scatter(norm * h1) + b1
    gcn_gemm_wmma<false><<<nb_gemm, 256, 0, stream>>>(x, W1, hbuf, N_NODES);
    agg_init_kernel<<<nb_init, 256, 0, stream>>>(hbuf, dinv, b1, abuf);
    agg_edges_kernel<<<nb_scatter, 256, 0, stream>>>(ei, dinv, hbuf, abuf);

    // layer 2: h2 = relu(out1) @ W2 ; out = scatter(norm * h2) + b2
    gcn_gemm_wmma<true><<<nb_gemm, 256, 0, stream>>>(abuf, W2, hbuf, N_NODES);
    agg_init_kernel<<<nb_init, 256, 0, stream>>>(hbuf, dinv, b2, out);
    agg_edges_kernel<<<nb_scatter, 256, 0, stream>>>(ei, dinv, hbuf, out);
}